// GATClassifier_10325101379600
// MI455X (gfx1250) — compile-verified
//
#include <hip/hip_runtime.h>

#define NN   50000
#define EE   800000
#define HH   8
#define DD   16
#define DHID 256
#define HD   128   // H*D

typedef float v2f __attribute__((ext_vector_type(2)));
typedef float v8f __attribute__((ext_vector_type(8)));

// Monotonic float<->uint encoding so float-max == uint-max.
__device__ __forceinline__ unsigned enc_f32(float f) {
  unsigned u = __float_as_uint(f);
  return (u & 0x80000000u) ? ~u : (u | 0x80000000u);
}
__device__ __forceinline__ float dec_f32(unsigned e) {
  unsigned u = (e & 0x80000000u) ? (e ^ 0x80000000u) : ~e;
  return __uint_as_float(u);
}
__device__ __forceinline__ float lrelu(float v, float s) { return v > 0.f ? v : s * v; }

// ---------------- init: zero rst0,rst1,s0,s1,m0,m1 (contiguous tail) ----------------
__global__ void k_zero(float* __restrict__ p, int n) {
  int i = blockIdx.x * blockDim.x + threadIdx.x;
  if (i < n) p[i] = 0.0f;
}

// ---------------- h = feat @ W via V_WMMA_F32_16X16X4_F32 ----------------
// grid = (3125, 8, 2), block = 32 (one wave per 16x16 output tile)
__global__ __launch_bounds__(32) void k_gemm_wmma(
    const float* __restrict__ feat, const float* __restrict__ W0,
    const float* __restrict__ W1, float* __restrict__ h0, float* __restrict__ h1) {
  const int mt   = blockIdx.x;            // row tile (16 rows)
  const int nt   = blockIdx.y;            // col tile (16 cols of 128)
  const float* W = blockIdx.z ? W1 : W0;
  float* Hout    = blockIdx.z ? h1 : h0;

  const int lane = threadIdx.x;
  const int half = lane >> 4;             // 0: K pair {0,1}, 1: K pair {2,3}
  const int l16  = lane & 15;

  const float* arow = feat + (mt * 16 + l16) * DHID;   // A row for this lane
  const int    col  = nt * 16 + l16;                   // B/C column for this lane

  v8f c = {};
  for (int k = 0; k < DHID; k += 4) {
    const int ka = k + 2 * half;
    // A 16x4: VGPR0 = K=ka, VGPR1 = K=ka+1 (contiguous in the feat row)
    v2f a = *(const v2f*)(arow + ka);
    // B 4x16: lane = column, VGPR pair holds rows K=ka, ka+1
    v2f b;
    b.x = W[ka * HD + col];
    b.y = W[(ka + 1) * HD + col];
    c = __builtin_amdgcn_wmma_f32_16x16x4_f32(false, a, false, b,
                                              (short)0, c, false, false);
  }
  // C/D layout: VGPR v -> row (v + 8*half), col l16
  float* out = Hout + (mt * 16 + 8 * half) * HD + nt * 16 + l16;
#pragma unroll
  for (int v = 0; v < 8; ++v) out[v * HD] = c[v];
}

// ---------------- el/er = reduce_d(h * a{l,r}) ----------------
__global__ void k_eler(const float* __restrict__ h0, const float* __restrict__ h1,
                       const float* __restrict__ al0, const float* __restrict__ ar0,
                       const float* __restrict__ al1, const float* __restrict__ ar1,
                       float* __restrict__ el0, float* __restrict__ er0,
                       float* __restrict__ el1, float* __restrict__ er1) {
  int gid = blockIdx.x * blockDim.x + threadIdx.x;   // over 2*N*H
  if (gid >= 2 * NN * HH) return;
  const int rel = gid >= NN * HH;
  const int nh  = rel ? gid - NN * HH : gid;
  const int n = nh >> 3, hd = nh & 7;
  const float* hp = (rel ? h1 : h0) + n * HD + hd * DD;
  const float* al = (rel ? al1 : al0) + hd * DD;
  const float* ar = (rel ? ar1 : ar0) + hd * DD;
  float sl = 0.f, sr = 0.f;
#pragma unroll
  for (int d = 0; d < DD; ++d) { sl += hp[d] * al[d]; sr += hp[d] * ar[d]; }
  (rel ? el1 : el0)[nh] = sl;
  (rel ? er1 : er0)[nh] = sr;
}

// ---------------- pass A: segment max (encoded uint atomicMax) ----------------
__global__ void k_edge_max(const int* __restrict__ src, const int* __restrict__ dst,
                           const float* __restrict__ el, const float* __restrict__ er,
                           unsigned* __restrict__ m) {
  int gid = blockIdx.x * blockDim.x + threadIdx.x;   // E*H
  if (gid >= EE * HH) return;
  const int e = gid >> 3, hd = gid & 7;
  const int s = src[e], d = dst[e];
  float v = lrelu(el[s * HH + hd] + er[d * HH + hd], 0.2f);
  atomicMax(m + d * HH + hd, enc_f32(v));
}

// ---------------- pass B: segment sum of exp(e - m[dst]) ----------------
__global__ void k_edge_expsum(const int* __restrict__ src, const int* __restrict__ dst,
                              const float* __restrict__ el, const float* __restrict__ er,
                              const unsigned* __restrict__ m, float* __restrict__ ssum) {
  int gid = blockIdx.x * blockDim.x + threadIdx.x;   // E*H
  if (gid >= EE * HH) return;
  const int e = gid >> 3, hd = gid & 7;
  const int s = src[e], d = dst[e];
  float v  = lrelu(el[s * HH + hd] + er[d * HH + hd], 0.2f);
  float ex = __expf(v - dec_f32(m[d * HH + hd]));
  atomicAdd(ssum + d * HH + hd, ex);
}

// ---------------- pass C: rst[dst] += alpha * h[src]  (L2-resident atomics) ----------------
__global__ void k_scatter(const float* __restrict__ hmat,
                          const int* __restrict__ src, const int* __restrict__ dst,
                          const float* __restrict__ el, const float* __restrict__ er,
                          const unsigned* __restrict__ m, const float* __restrict__ ssum,
                          float* __restrict__ rst) {
  int gid = blockIdx.x * blockDim.x + threadIdx.x;   // E*128
  if (gid >= EE * HD) return;
  const int e = gid >> 7, c = gid & 127, hd = c >> 4;
  const int s = src[e], d = dst[e];
  float v     = lrelu(el[s * HH + hd] + er[d * HH + hd], 0.2f);
  float alpha = __expf(v - dec_f32(m[d * HH + hd])) / ssum[d * HH + hd];
  atomicAdd(rst + d * HD + c, alpha * hmat[s * HD + c]);
}

// ---------------- finalize: out[n,d] = (1/D) * sum_h 0.5*(lr(rst0+b0)+lr(rst1+b1)) ----------------
__global__ void k_final(const float* __restrict__ rst0, const float* __restrict__ rst1,
                        const float* __restrict__ b0, const float* __restrict__ b1,
                        float* __restrict__ out) {
  int gid = blockIdx.x * blockDim.x + threadIdx.x;   // N*D
  if (gid >= NN * DD) return;
  const int n = gid >> 4, d = gid & 15;
  float acc = 0.f;
#pragma unroll
  for (int hd = 0; hd < HH; ++hd) {
    float v0 = lrelu(rst0[n * HD + hd * DD + d] + b0[hd * DD + d], 0.1f);
    float v1 = lrelu(rst1[n * HD + hd * DD + d] + b1[hd * DD + d], 0.1f);
    acc += 0.5f * (v0 + v1);
  }
  out[gid] = acc * (1.0f / (float)DD);
}

extern "C" void kernel_launch(void* const* d_in, const int* in_sizes, int n_in,
                              void* d_out, int out_size, void* d_ws, size_t ws_size,
                              hipStream_t stream) {
  const float* feat = (const float*)d_in[0];
  const int*   src0 = (const int*)d_in[1];
  const int*   dst0 = (const int*)d_in[2];
  const int*   src1 = (const int*)d_in[3];
  const int*   dst1 = (const int*)d_in[4];
  const float* W0   = (const float*)d_in[5];
  const float* al0  = (const float*)d_in[6];
  const float* ar0  = (const float*)d_in[7];
  const float* b0   = (const float*)d_in[8];
  const float* W1   = (const float*)d_in[9];
  const float* al1  = (const float*)d_in[10];
  const float* ar1  = (const float*)d_in[11];
  const float* b1   = (const float*)d_in[12];
  float* out = (float*)d_out;

  // Workspace carve-up (floats): h0,h1 | el0,er0,el1,er1 | rst0,rst1,s0,s1,m0,m1
  float* h0   = (float*)d_ws;
  float* h1   = h0 + (size_t)NN * HD;
  float* el0  = h1 + (size_t)NN * HD;
  float* er0  = el0 + (size_t)NN * HH;
  float* el1  = er0 + (size_t)NN * HH;
  float* er1  = el1 + (size_t)NN * HH;
  float* rst0 = er1 + (size_t)NN * HH;     // ---- zeroed region starts here ----
  float* rst1 = rst0 + (size_t)NN * HD;
  float* s0   = rst1 + (size_t)NN * HD;
  float* s1   = s0 + (size_t)NN * HH;
  unsigned* m0 = (unsigned*)(s1 + (size_t)NN * HH);
  unsigned* m1 = m0 + (size_t)NN * HH;

  const int zeroCount = 2 * NN * HD + 4 * NN * HH;   // rst0,rst1,s0,s1,m0,m1
  k_zero<<<(zeroCount + 255) / 256, 256, 0, stream>>>(rst0, zeroCount);

  // Projection GEMMs (WMMA f32 16x16x4): 3125 row tiles x 8 col tiles x 2 relations
  k_gemm_wmma<<<dim3(NN / 16, HD / 16, 2), 32, 0, stream>>>(feat, W0, W1, h0, h1);

  // Per-node attention logits
  k_eler<<<(2 * NN * HH + 255) / 256, 256, 0, stream>>>(h0, h1, al0, ar0, al1, ar1,
                                                        el0, er0, el1, er1);

  const int nEH = EE * HH;
  k_edge_max<<<(nEH + 255) / 256, 256, 0, stream>>>(src0, dst0, el0, er0, m0);
  k_edge_max<<<(nEH + 255) / 256, 256, 0, stream>>>(src1, dst1, el1, er1, m1);

  k_edge_expsum<<<(nEH + 255) / 256, 256, 0, stream>>>(src0, dst0, el0, er0, m0, s0);
  k_edge_expsum<<<(nEH + 255) / 256, 256, 0, stream>>>(src1, dst1, el1, er1, m1, s1);

  const int nEC = EE * HD;
  k_scatter<<<(nEC + 255) / 256, 256, 0, stream>>>(h0, src0, dst0, el0, er0, m0, s0, rst0);
  k_scatter<<<(nEC + 255) / 256, 256, 0, stream>>>(h1, src1, dst1, el1, er1, m1, s1, rst1);

  k_final<<<(NN * DD + 255) / 256, 256, 0, stream>>>(rst0, rst1, b0, b1, out);
}